// TimeDomainModel_38508676776433
// MI455X (gfx1250) — compile-verified
//
#include <hip/hip_runtime.h>
#include <hip/hip_bf16.h>

// Balloon-Windkessel BOLD, explicit Euler, 1000 steps — single wave32,
// pure-VALU, chain-minimized, deferred readout.
//
// Reference constants (kappa = gamma = 1, u_t = 1 baked in):
//   DT=0.01, 1/ALPHA=3.125 (exact), RHO=0.34, 1/RHO=2.9411765
//   log2(1-RHO)=log2(0.66)=-0.59946167
//   V0*K1=0.0476, V0*K2=0.04, V0*K3=0.0096, C0=0.0972
//
// Per-step recurrence (exact reference algebra, fma-collapsed):
//   dtim  = DT/mtt, Cv = log2(dtim), B = dtim/RHO          (uniform, hoisted)
//   vpowC = dtim * v^(1/a) = exp2(3.125*log2(v) + Cv)      (post-exp mul folded)
//   fE    = f*E(f)         = fma(-f, exp2(log2(.66)/f), f)
//   s'    = fma(0.99, s, fma(-0.01, f, 0.02))              ; s + DT*(2-s-f)
//   f'    = fma(0.01, s_old, f)
//   v'    = fma(dtim, f, v) - vpowC
//   q'    = fma(B, fE, fma(-vpowC, rv*q, q))               ; rv = 1/v carried
//   rv'   = rcp(v')                                         ; feeds next dq
//
// Readout deferral: lane L latches (q',v') at steps with (t%32)==L (two
// cndmasks); y = 0.0972 - 0.0476*q - 0.04*q/v - 0.0096*v is computed ONCE per
// 32-step group (one rcp + 4 ops) and stored as a coalesced 128B wave store.
// 1000 = 31*32 + 8 (tail handled). State pinned in VGPRs via opaque zero so
// the compiler cannot scalarize the serial chain onto SALU.

__global__ __launch_bounds__(32) void bold_euler_kernel(const float* __restrict__ mtt_p,
                                                        float* __restrict__ out) {
    const int lane = (int)threadIdx.x;   // wave32 lane id 0..31

    const float mtt     = mtt_p[0];
    const float inv_mtt = 1.0f / mtt;                       // one-time precise divide
    const float dtim    = 0.01f * inv_mtt;                  // DT/mtt
    const float Cv      = __builtin_amdgcn_logf(dtim);      // log2(DT/mtt), one-time
    const float B       = dtim * 2.9411765f;                // DT/(mtt*RHO)

    // Opaque VGPR zero: exact identity, defeats uniformity analysis so the
    // recurrence stays on the VALU/TRANS pipes (no SALU ping-pong).
    float z;
    asm("v_mov_b32 %0, 0" : "=v"(z));

    float s  = z;
    float f  = 1.0f + z;
    float v  = 1.0f + z;
    float q  = 1.0f + z;
    float rv = 1.0f + z;                 // carried 1/v (v == 1 initially)

    float qL = z, vL = 1.0f + z;         // this lane's latched post-update state

    auto step = [&](int r) {
        // --- critical v-chain: log2 -> fma -> exp2 ---
        const float log2v = __builtin_amdgcn_logf(v);                        // v_log_f32
        const float vpowC = __builtin_amdgcn_exp2f(fmaf(3.125f, log2v, Cv)); // v_exp_f32

        // --- f-side (short chain, overlaps with v-chain) ---
        const float rf  = __builtin_amdgcn_rcpf(f);                          // v_rcp_f32
        const float e66 = __builtin_amdgcn_exp2f(-0.59946167f * rf);         // 0.66^(1/f)
        const float fE  = fmaf(-f, e66, f);                                  // f*E(f)

        // --- independent partials (overlap TRANS latency) ---
        const float vtmp = fmaf(dtim, f, v);         // v + (DT/mtt)*f
        const float rvq  = rv * q;                   // q/v (old state)
        const float qtmp = fmaf(-vpowC, rvq, q);

        // --- Euler update (old-state semantics preserved) ---
        const float s_old = s;
        s = fmaf(0.99f, s, fmaf(-0.01f, f, 0.02f));  // s + DT*(2 - s - f)
        f = fmaf(0.01f, s_old, f);                   // f + DT*s_old
        v = vtmp - vpowC;                            // v + DT*(f - v^(1/a))/mtt
        q = fmaf(B, fE, qtmp);                       // q + DT*(fE/rho - vpow*q/v)/mtt

        rv = __builtin_amdgcn_rcpf(v);               // v_rcp_f32; next dq

        if (r == lane) { qL = q; vL = v; }           // two v_cndmask latches
    };

    // y from latched post-update state (once per group, off the hot loop)
    auto readout = [&]() -> float {
        const float rvL = __builtin_amdgcn_rcpf(vL);
        float y = fmaf(-0.0476f, qL, 0.0972f);
        y = fmaf(-0.04f,  qL * rvL, y);
        y = fmaf(-0.0096f, vL, y);
        return y;
    };

    // 31 full groups of 32 steps -> one coalesced 128B wave store per group
    for (int g = 0; g < 31; ++g) {
        #pragma unroll
        for (int r = 0; r < 32; ++r) {
            step(r);
        }
        out[g * 32 + lane] = readout();
    }

    // Tail: steps 992..999, lanes 0..7 hold valid data
    #pragma unroll
    for (int r = 0; r < 8; ++r) {
        step(r);
    }
    if (lane < 8) {
        out[992 + lane] = readout();
    }
}

extern "C" void kernel_launch(void* const* d_in, const int* in_sizes, int n_in,
                              void* d_out, int out_size, void* d_ws, size_t ws_size,
                              hipStream_t stream) {
    (void)in_sizes; (void)n_in; (void)out_size; (void)d_ws; (void)ws_size;
    const float* mtt = (const float*)d_in[0];   // single-element f32 parameter
    float* out = (float*)d_out;                 // 1000 f32 outputs
    // One wave32 performs the entire (strictly serial) integration.
    bold_euler_kernel<<<1, 32, 0, stream>>>(mtt, out);
}